// BBoxAnchors_19868518711895
// MI455X (gfx1250) — compile-verified
//
#include <hip/hip_runtime.h>
#include <stdint.h>

#define A_TOTAL 43008
#define MBOX    128
#define BATCH   16
#define IOU_THR 0.3f
#define BPB     4        // boxes per block in box_max_kernel

// ---------------------------------------------------------------------------
// IoU of one anchor (xyxy + area) vs one box (xyxy + area)
// ---------------------------------------------------------------------------
__device__ __forceinline__ float iou_pair(float ax1, float ay1, float ax2, float ay2, float areaA,
                                          float bx1, float by1, float bx2, float by2, float areaB) {
    float lx = fmaxf(ax1, bx1);
    float ly = fmaxf(ay1, by1);
    float rx = fminf(ax2, bx2);
    float ry = fminf(ay2, by2);
    float w  = fmaxf(rx - lx, 0.0f);
    float h  = fmaxf(ry - ly, 0.0f);
    float inter = w * h;
    return inter / (areaA + areaB - inter);
}

// ---------------------------------------------------------------------------
// Kernel 1: per-anchor max IoU + argmax over the 128 GT boxes of one batch.
// GT boxes (2KB) staged into LDS with CDNA5 async global->LDS b128 loads.
// grid = (A/256, B), block = 256 (8 wave32)
// ---------------------------------------------------------------------------
__global__ void __launch_bounds__(256)
anchor_max_kernel(const float* __restrict__ bboxes,   // [B,M,4] xyxy
                  const float* __restrict__ anchors,  // [A,4] cxcywh
                  float* __restrict__ ws_mioa,        // [B,A]
                  int*   __restrict__ ws_bi) {        // [B,A]
    __shared__ float4 s_box[MBOX];
    __shared__ float  s_area[MBOX];

    const int tid = threadIdx.x;
    const int b   = blockIdx.y;
    const float* gbase = bboxes + (size_t)b * MBOX * 4;

    if (tid < MBOX) {
        // LDS byte address = low 32 bits of the flat shared-aperture address
        unsigned lds_addr = (unsigned)(size_t)(&s_box[tid]);
        unsigned goff     = (unsigned)(tid * 16);  // one b128 (one box) per lane
        asm volatile("global_load_async_to_lds_b128 %0, %1, %2"
                     :: "v"(lds_addr), "v"(goff), "s"(gbase)
                     : "memory");
    }
    asm volatile("s_wait_asynccnt 0x0" ::: "memory");
    __syncthreads();

    if (tid < MBOX) {
        float4 bx = s_box[tid];
        s_area[tid] = (bx.z - bx.x) * (bx.w - bx.y);
    }
    __syncthreads();

    const int a = blockIdx.x * 256 + tid;
    float4 anc = ((const float4*)anchors)[a];          // (cx, cy, w, h)
    float aw = anc.z, ah = anc.w;
    float ax1 = anc.x - 0.5f * aw, ay1 = anc.y - 0.5f * ah;
    float ax2 = anc.x + 0.5f * aw, ay2 = anc.y + 0.5f * ah;
    float areaA = aw * ah;

    float best = -1.0f;  // IoU >= 0, so j=0 always wins first -> first-occurrence argmax
    int   bestj = 0;
#pragma unroll 4
    for (int j = 0; j < MBOX; ++j) {
        float4 bx = s_box[j];
        float v = iou_pair(ax1, ay1, ax2, ay2, areaA, bx.x, bx.y, bx.z, bx.w, s_area[j]);
        if (v > best) { best = v; bestj = j; }   // strict > keeps first occurrence
    }
    size_t idx = (size_t)b * A_TOTAL + a;
    ws_mioa[idx] = best;
    ws_bi[idx]   = bestj;
}

// ---------------------------------------------------------------------------
// Kernel 2: per-box max IoU + argmax over all 43008 anchors.
// One block per (batch, 4 boxes): each anchor is loaded once and scored
// against 4 block-uniform boxes (4x less L2 anchor traffic, conversion
// amortized). Deterministic packed-u64 LDS tree reduction; pack =
// (iou_bits << 32) | ~anchor -> equal IoU picks smaller anchor index
// (JAX first-occurrence argmax), no atomics.
// grid = (M/BPB, B), block = 256
// ---------------------------------------------------------------------------
__global__ void __launch_bounds__(256)
box_max_kernel(const float* __restrict__ bboxes,   // [B,M,4] xyxy
               const float* __restrict__ anchors,  // [A,4] cxcywh
               float* __restrict__ ws_mb,          // [B,M]
               int*   __restrict__ ws_ai) {        // [B,M]
    __shared__ unsigned long long s_red[256];

    const int tid = threadIdx.x;
    const int j0  = blockIdx.x * BPB;
    const int b   = blockIdx.y;

    // 4 block-uniform boxes -> scalar loads
    float bx1[BPB], by1[BPB], bx2[BPB], by2[BPB], areaB[BPB];
#pragma unroll
    for (int r = 0; r < BPB; ++r) {
        const float* bb = bboxes + ((size_t)b * MBOX + j0 + r) * 4;
        bx1[r] = bb[0]; by1[r] = bb[1]; bx2[r] = bb[2]; by2[r] = bb[3];
        areaB[r] = (bx2[r] - bx1[r]) * (by2[r] - by1[r]);
    }

    float best[BPB];
    int   bestA[BPB];
#pragma unroll
    for (int r = 0; r < BPB; ++r) { best[r] = -1.0f; bestA[r] = 0; }

    for (int a = tid; a < A_TOTAL; a += 256) {   // ascending a -> strict > keeps smallest a
        float4 anc = ((const float4*)anchors)[a];
        float aw = anc.z, ah = anc.w;
        float ax1 = anc.x - 0.5f * aw, ay1 = anc.y - 0.5f * ah;
        float ax2 = anc.x + 0.5f * aw, ay2 = anc.y + 0.5f * ah;
        float areaA = aw * ah;
#pragma unroll
        for (int r = 0; r < BPB; ++r) {
            float v = iou_pair(ax1, ay1, ax2, ay2, areaA,
                               bx1[r], by1[r], bx2[r], by2[r], areaB[r]);
            if (v > best[r]) { best[r] = v; bestA[r] = a; }
        }
    }

#pragma unroll
    for (int r = 0; r < BPB; ++r) {
        unsigned long long pack =
            ((unsigned long long)__float_as_uint(best[r]) << 32) |
            (unsigned)(~(unsigned)bestA[r]);
        s_red[tid] = pack;
        __syncthreads();
#pragma unroll
        for (int s = 128; s > 0; s >>= 1) {
            if (tid < s) {
                unsigned long long x = s_red[tid], y = s_red[tid + s];
                s_red[tid] = (x >= y) ? x : y;
            }
            __syncthreads();
        }
        if (tid == 0) {
            unsigned long long p = s_red[0];
            ws_mb[b * MBOX + j0 + r] = __uint_as_float((unsigned)(p >> 32));
            ws_ai[b * MBOX + j0 + r] = (int)(~(unsigned)p);
        }
        __syncthreads();   // protect s_red[0] read before next-round overwrite
    }
}

// ---------------------------------------------------------------------------
// Kernel 3: sequential-override replay + scoring + matched-box gather.
// For each anchor, scan j ascending: if anchor_indice[j]==a, override with
// (max_iou_of_bbox[j], j) -> last (largest) j wins, identical to the scan.
// grid = (A/256, B), block = 256
// ---------------------------------------------------------------------------
__global__ void __launch_bounds__(256)
finalize_kernel(const int*   __restrict__ labels,   // [B,M]
                const float* __restrict__ bboxes,   // [B,M,4]
                const float* __restrict__ ws_mioa,  // [B,A]
                const int*   __restrict__ ws_bi,    // [B,A]
                const float* __restrict__ ws_mb,    // [B,M]
                const int*   __restrict__ ws_ai,    // [B,M]
                float* __restrict__ out_scores,     // [B,A]
                float* __restrict__ out_matched) {  // [B,A,4]
    __shared__ float4 s_box[MBOX];
    __shared__ float  s_mb[MBOX];
    __shared__ int    s_ai[MBOX];
    __shared__ int    s_lab[MBOX];

    const int tid = threadIdx.x;
    const int b   = blockIdx.y;

    if (tid < MBOX) {
        // stage the matched-box tile through the async global->LDS path
        const float* gbase = bboxes + (size_t)b * MBOX * 4;
        unsigned lds_addr = (unsigned)(size_t)(&s_box[tid]);
        unsigned goff     = (unsigned)(tid * 16);
        asm volatile("global_load_async_to_lds_b128 %0, %1, %2"
                     :: "v"(lds_addr), "v"(goff), "s"(gbase)
                     : "memory");
        s_mb[tid]  = ws_mb[b * MBOX + tid];
        s_ai[tid]  = ws_ai[b * MBOX + tid];
        s_lab[tid] = labels[b * MBOX + tid];
    }
    asm volatile("s_wait_asynccnt 0x0" ::: "memory");
    __syncthreads();

    const int a = blockIdx.x * 256 + tid;
    size_t idx = (size_t)b * A_TOTAL + a;
    float mioa = ws_mioa[idx];
    int   bi   = ws_bi[idx];

#pragma unroll 8
    for (int j = 0; j < MBOX; ++j) {
        if (s_ai[j] == a) { mioa = s_mb[j]; bi = j; }   // ascending j: later GT wins
    }

    float denom = fmaxf(s_mb[bi], IOU_THR);
    mioa = (mioa < IOU_THR * 0.5f) ? 0.0f : mioa;
    float sc = mioa / denom;
    if (s_lab[bi] <= 0) sc = 0.0f;

    out_scores[idx] = sc;
    ((float4*)out_matched)[idx] = s_box[bi];
}

// ---------------------------------------------------------------------------
extern "C" void kernel_launch(void* const* d_in, const int* in_sizes, int n_in,
                              void* d_out, int out_size, void* d_ws, size_t ws_size,
                              hipStream_t stream) {
    const int*   labels  = (const int*)d_in[0];    // [16,128] int32
    const float* bboxes  = (const float*)d_in[1];  // [16,128,4] f32 xyxy
    const float* anchors = (const float*)d_in[2];  // [43008,4] f32 cxcywh

    float* out = (float*)d_out;                    // scores [B*A] ++ matched [B*A*4]

    char* ws = (char*)d_ws;                        // ~5.5 MB total, fully overwritten
    float* ws_mioa = (float*)ws;                                        // B*A f32
    int*   ws_bi   = (int*)  (ws + (size_t)BATCH * A_TOTAL * 4);        // B*A i32
    float* ws_mb   = (float*)(ws + (size_t)BATCH * A_TOTAL * 8);        // B*M f32
    int*   ws_ai   = (int*)  (ws + (size_t)BATCH * A_TOTAL * 8 + (size_t)BATCH * MBOX * 4);

    dim3 blk(256);
    anchor_max_kernel<<<dim3(A_TOTAL / 256, BATCH), blk, 0, stream>>>(bboxes, anchors, ws_mioa, ws_bi);
    box_max_kernel<<<dim3(MBOX / BPB, BATCH), blk, 0, stream>>>(bboxes, anchors, ws_mb, ws_ai);
    finalize_kernel<<<dim3(A_TOTAL / 256, BATCH), blk, 0, stream>>>(labels, bboxes,
                                                                    ws_mioa, ws_bi, ws_mb, ws_ai,
                                                                    out, out + (size_t)BATCH * A_TOTAL);
}